// DictLearn_21904333209669
// MI455X (gfx1250) — compile-verified
//
#include <hip/hip_runtime.h>
#include <math.h>

// Problem constants (from reference)
#define K_ATOMS   512      // NUM_EMB
#define DCH       64       // EMB_DIM
#define SPARS     8        // SPARSITY
#define NSIG      32768    // 32*32*32 signals
#define RECON_N   2097152  // 32*64*32*32
#define PERP_OFF  2097153
#define GAMMA_OFF 2097154
#define NGROUPS   2048     // NSIG / 16

typedef float v2f __attribute__((ext_vector_type(2)));
typedef float v8f __attribute__((ext_vector_type(8)));
typedef int   b128_t __attribute__((vector_size(16)));  // int4 for async-LDS ops

#define AS1 __attribute__((address_space(1)))
#define AS3 __attribute__((address_space(3)))

#if defined(__has_builtin)
#if __has_builtin(__builtin_amdgcn_global_load_async_to_lds_b128)
#define HAVE_ASYNC_LDS 1
#endif
#endif

// D = A(16x4 f32) * B(4x16 f32) + C  -- CDNA5 V_WMMA_F32_16X16X4_F32
__device__ __forceinline__ v8f wmma16x16x4(v2f a, v2f b, v8f c) {
  return __builtin_amdgcn_wmma_f32_16x16x4_f32(false, a, false, b, (short)0, c,
                                               false, false);
}

// ---------------------------------------------------------------------------
// K0: G = dict^T * dict  (512x512), dict is (64,512) row-major.
// One wave computes one 16x16 tile; K-loop = 64/4 = 16 WMMA ops.
// ---------------------------------------------------------------------------
__global__ __launch_bounds__(128) void gram_kernel(
    const float* __restrict__ dict, float* __restrict__ G) {
  int wave = threadIdx.x >> 5;
  int lane = threadIdx.x & 31;
  int l15  = lane & 15;
  int hiK  = (lane >> 4) << 1;  // 0 for lanes 0-15, 2 for lanes 16-31
  int tile = blockIdx.x * 4 + wave;       // 0..1023
  int tm = (tile >> 5) * 16;              // 32 tiles per dim
  int tn = (tile & 31) * 16;
  v8f acc = {};
#pragma unroll
  for (int kk = 0; kk < 16; ++kk) {
    int k0 = kk * 4 + hiK;
    v2f a, b;
    a.x = dict[(k0 + 0) * K_ATOMS + tm + l15];  // A[m][k] = dict[k][m]
    a.y = dict[(k0 + 1) * K_ATOMS + tm + l15];
    b.x = dict[(k0 + 0) * K_ATOMS + tn + l15];  // B[k][n] = dict[k][n]
    b.y = dict[(k0 + 1) * K_ATOMS + tn + l15];
    acc = wmma16x16x4(a, b, acc);
  }
#pragma unroll
  for (int r = 0; r < 8; ++r) {
    int M = r + ((lane >> 4) << 3);
    G[(tm + M) * K_ATOMS + tn + l15] = acc[r];
  }
}

// ---------------------------------------------------------------------------
// K1: gather z_e (B,C,H,W) into xT (NSIG x 64) signal-major.
// x[c][n] (reference's raw reshape) lives at z_e flat index decoded from
// i = c*32768 + n  ->  (b,h,w,c') of the (B,H,W,C) permutation.
// ---------------------------------------------------------------------------
__global__ void permute_kernel(const float* __restrict__ z_e,
                               float* __restrict__ xT) {
  int idx = blockIdx.x * blockDim.x + threadIdx.x;  // 0..2M-1
  int n = idx >> 6, c = idx & 63;
  unsigned i = ((unsigned)c << 15) + (unsigned)n;
  unsigned b = i >> 16, rem = i & 65535u;
  unsigned h = rem >> 11, r2 = rem & 2047u;
  unsigned w = r2 >> 6, cc = r2 & 63u;
  xT[idx] = z_e[(b << 16) + (cc << 10) + (h << 5) + w];
}

// ---------------------------------------------------------------------------
// K2: zero the dense gamma output region (scatter fills 8/512 per column).
// ---------------------------------------------------------------------------
__global__ void zero_gamma_kernel(float* __restrict__ gout) {
  size_t i = (size_t)blockIdx.x * blockDim.x + threadIdx.x;  // float2 index
  ((float2*)gout)[i] = make_float2(0.f, 0.f);
}

// ---------------------------------------------------------------------------
// K3: fused batch-OMP. 1 workgroup (4 waves) = 16 signals.
// Phase A: stage xT block to LDS with async-copy, then
//          corr_init (16x512) via WMMA f32 16x16x4 into LDS.
// Phase B: each wave runs full OMP on 4 signals (argmax = wave reduction,
//          Cholesky lane-replicated + fully unrolled so L stays in VGPRs).
// Phase C: gamma scatter, recon + sq-error, entropy partials.
// ---------------------------------------------------------------------------
__global__ __launch_bounds__(128) void omp_kernel(
    const float* __restrict__ xT, const float* __restrict__ dict,
    const float* __restrict__ G, float* __restrict__ out,
    float* __restrict__ hPart, float* __restrict__ sqPart) {
  __shared__ float    corrI[16][K_ATOMS];            // 32 KB
  __shared__ __align__(16) float xStage[16 * DCH];   // 4 KB, signal-major
  __shared__ unsigned omega[16][16];                 // 512-bit mask / signal
  __shared__ float    waveH[4], waveSq[4];

  int wave = threadIdx.x >> 5;
  int lane = threadIdx.x & 31;
  int l15  = lane & 15;
  int hiK  = (lane >> 4) << 1;
  int group = blockIdx.x;
  int n0 = group * 16;

  // ---- Stage the 16x64 xT block (contiguous 4 KB) into LDS
  {
    const float* gsrc = xT + (size_t)n0 * DCH;
#if defined(HAVE_ASYNC_LDS)
    // CDNA5 async global->LDS copy, tracked by ASYNCcnt
    __builtin_amdgcn_global_load_async_to_lds_b128(
        (AS1 b128_t*)(gsrc + threadIdx.x * 4),
        (AS3 b128_t*)(xStage + threadIdx.x * 4), 0, 0);
    __builtin_amdgcn_global_load_async_to_lds_b128(
        (AS1 b128_t*)(gsrc + 512 + threadIdx.x * 4),
        (AS3 b128_t*)(xStage + 512 + threadIdx.x * 4), 0, 0);
#if __has_builtin(__builtin_amdgcn_s_wait_asynccnt)
    __builtin_amdgcn_s_wait_asynccnt(0);
#else
    asm volatile("s_wait_asynccnt 0x0" ::: "memory");
#endif
#else
    ((float4*)xStage)[threadIdx.x] = ((const float4*)gsrc)[threadIdx.x];
    ((float4*)xStage)[threadIdx.x + 128] =
        ((const float4*)gsrc)[threadIdx.x + 128];
#endif
  }
  ((unsigned*)omega)[threadIdx.x]       = 0xFFFFFFFFu;
  ((unsigned*)omega)[threadIdx.x + 128] = 0xFFFFFFFFu;
  __syncthreads();

  // ---- Phase A: corr_init = xT(16x64) * dict(64x512)
  v2f afrag[16];
#pragma unroll
  for (int kk = 0; kk < 16; ++kk) {
    afrag[kk].x = xStage[l15 * DCH + kk * 4 + hiK + 0];
    afrag[kk].y = xStage[l15 * DCH + kk * 4 + hiK + 1];
  }
#pragma unroll
  for (int t = 0; t < 8; ++t) {
    int tn = (wave * 8 + t) * 16;
    v8f acc = {};
#pragma unroll
    for (int kk = 0; kk < 16; ++kk) {
      v2f b;
      b.x = dict[(kk * 4 + hiK + 0) * K_ATOMS + tn + l15];
      b.y = dict[(kk * 4 + hiK + 1) * K_ATOMS + tn + l15];
      acc = wmma16x16x4(afrag[kk], b, acc);
    }
#pragma unroll
    for (int r = 0; r < 8; ++r) {
      int M = r + ((lane >> 4) << 3);
      corrI[M][tn + l15] = acc[r];
    }
  }
  __syncthreads();

  // ---- Phase B/C: per-wave OMP over 4 signals
  float hAcc = 0.f, sqAcc = 0.f;
  for (int q = 0; q < 4; ++q) {
    int s = wave * 4 + q;
    int n = n0 + s;
    int   I[SPARS];
    float g[SPARS];
    float L[SPARS][SPARS];
#pragma unroll
    for (int a = 0; a < SPARS; ++a) {
      I[a] = 0; g[a] = 0.f;
#pragma unroll
      for (int bb = 0; bb < SPARS; ++bb) L[a][bb] = 0.f;
    }

#pragma unroll
    for (int k = 1; k <= SPARS; ++k) {
      // argmax over where(omega, |corr_init - sum_j g_j G[I_j]|, 0)
      float best = -1.f;
      int   bidx = 0;
#pragma unroll
      for (int ch = 0; ch < 16; ++ch) {
        int m = ch * 32 + lane;
        float cv = corrI[s][m];
#pragma unroll
        for (int j = 0; j < SPARS; ++j)
          if (j < k - 1) cv -= g[j] * G[I[j] * K_ATOMS + m];
        unsigned wb = omega[s][ch];
        float v = ((wb >> lane) & 1u) ? fabsf(cv) : 0.f;
        if (v > best || (v == best && m < bidx)) { best = v; bidx = m; }
      }
      for (int off = 16; off > 0; off >>= 1) {
        float ov = __shfl_xor(best, off, 32);
        int   oi = __shfl_xor(bidx, off, 32);
        if (ov > best || (ov == best && oi < bidx)) { best = ov; bidx = oi; }
      }
      int khat = bidx;
      if (lane == 0) omega[s][khat >> 5] &= ~(1u << (khat & 31));

      // Cholesky rank-1 growth (lane-replicated scalar work)
      if (k > 1) {
        float w_[SPARS - 1];
#pragma unroll
        for (int j = 0; j < SPARS - 1; ++j) {
          float bj = 0.f;
          if (j < k - 1) {
            bj = G[I[j] * K_ATOMS + khat];
#pragma unroll
            for (int p = 0; p < SPARS - 1; ++p)
              if (p < j) bj -= L[j][p] * w_[p];
            bj /= L[j][j];
          }
          w_[j] = bj;
        }
        float ss = 0.f;
#pragma unroll
        for (int j = 0; j < SPARS - 1; ++j)
          if (j < k - 1) { ss += w_[j] * w_[j]; L[k - 1][j] = w_[j]; }
        L[k - 1][k - 1] = sqrtf(fmaxf(1.f - ss, 0.f));
      } else {
        L[0][0] = 1.f;
      }
      I[k - 1] = khat;

      // forward solve L y = corr_init[I], back solve L^T g = y
      float y[SPARS];
#pragma unroll
      for (int i2 = 0; i2 < SPARS; ++i2) {
        float bi = 0.f;
        if (i2 < k) {
          bi = corrI[s][I[i2]];
#pragma unroll
          for (int p = 0; p < SPARS; ++p)
            if (p < i2) bi -= L[i2][p] * y[p];
          bi /= L[i2][i2];
        }
        y[i2] = bi;
      }
#pragma unroll
      for (int i2 = SPARS - 1; i2 >= 0; --i2) {
        float bi = 0.f;
        if (i2 < k) {
          bi = y[i2];
#pragma unroll
          for (int p = 0; p < SPARS; ++p)
            if (p > i2 && p < k) bi -= L[p][i2] * g[p];
          bi /= L[i2][i2];
        }
        g[i2] = bi;
      }
    }

    // gamma scatter (region pre-zeroed)
    if (lane < SPARS)
      out[GAMMA_OFF + (size_t)I[lane] * NSIG + n] = g[lane];

    // entropy of softmax over 512 entries (504 zeros + 8 coeffs)
    {
      float mx = 0.f;
#pragma unroll
      for (int j = 0; j < SPARS; ++j) mx = fmaxf(mx, g[j]);
      float S = (float)(K_ATOMS - SPARS) * expf(-mx);
      float T = 0.f;
#pragma unroll
      for (int j = 0; j < SPARS; ++j) {
        float e = expf(g[j] - mx);
        S += e;
        T += g[j] * e;
      }
      hAcc += mx + logf(S) - T / S;
    }

    // recon + squared error; each lane covers channels lane and lane+32
    {
      float sq = 0.f;
#pragma unroll
      for (int half = 0; half < 2; ++half) {
        int c = lane + half * 32;
        float r = 0.f;
#pragma unroll
        for (int j = 0; j < SPARS; ++j) r += g[j] * dict[c * K_ATOMS + I[j]];
        float xv = xStage[s * DCH + c];
        float d = r - xv;
        sq += d * d;
        unsigned i = ((unsigned)c << 15) + (unsigned)n;
        unsigned b = i >> 16, rem = i & 65535u;
        unsigned h = rem >> 11, r2 = rem & 2047u;
        unsigned w = r2 >> 6, cc = r2 & 63u;
        out[1 + (b << 16) + (cc << 10) + (h << 5) + w] = r;
      }
      for (int off = 16; off > 0; off >>= 1) sq += __shfl_xor(sq, off, 32);
      sqAcc += sq;
    }
  }

  if (lane == 0) { waveH[wave] = hAcc; waveSq[wave] = sqAcc; }
  __syncthreads();
  if (threadIdx.x == 0) {
    hPart[group]  = waveH[0] + waveH[1] + waveH[2] + waveH[3];
    sqPart[group] = waveSq[0] + waveSq[1] + waveSq[2] + waveSq[3];
  }
}

// ---------------------------------------------------------------------------
// K4: deterministic fixed-order reduction -> loss, perplexity scalars.
// ---------------------------------------------------------------------------
__global__ __launch_bounds__(256) void finalize_kernel(
    const float* __restrict__ hPart, const float* __restrict__ sqPart,
    float* __restrict__ out) {
  __shared__ float sh[256], sq[256];
  float h = 0.f, s = 0.f;
  for (int i = threadIdx.x; i < NGROUPS; i += 256) {
    h += hPart[i];
    s += sqPart[i];
  }
  sh[threadIdx.x] = h;
  sq[threadIdx.x] = s;
  __syncthreads();
  for (int off = 128; off > 0; off >>= 1) {
    if ((int)threadIdx.x < off) {
      sh[threadIdx.x] += sh[threadIdx.x + off];
      sq[threadIdx.x] += sq[threadIdx.x + off];
    }
    __syncthreads();
  }
  if (threadIdx.x == 0) {
    out[0]        = 0.25f * sq[0] / (float)((size_t)DCH * NSIG);  // commit loss
    out[PERP_OFF] = expf(sh[0] / (float)NSIG);                    // perplexity
  }
}

extern "C" void kernel_launch(void* const* d_in, const int* in_sizes, int n_in,
                              void* d_out, int out_size, void* d_ws,
                              size_t ws_size, hipStream_t stream) {
  const float* z_e  = (const float*)d_in[0];  // (32,64,32,32)
  const float* dict = (const float*)d_in[1];  // (64,512)
  float* out = (float*)d_out;
  float* ws  = (float*)d_ws;

  float* xT     = ws;                 // 2,097,152 floats
  float* G      = xT + RECON_N;       //   262,144 floats
  float* hPart  = G + K_ATOMS * K_ATOMS;  // 2048
  float* sqPart = hPart + NGROUPS;        // 2048

  hipLaunchKernelGGL(gram_kernel, dim3(256), dim3(128), 0, stream, dict, G);
  hipLaunchKernelGGL(permute_kernel, dim3(RECON_N / 256), dim3(256), 0, stream,
                     z_e, xT);
  hipLaunchKernelGGL(zero_gamma_kernel, dim3((K_ATOMS * NSIG / 2) / 256),
                     dim3(256), 0, stream, out + GAMMA_OFF);
  hipLaunchKernelGGL(omp_kernel, dim3(NGROUPS), dim3(128), 0, stream, xT, dict,
                     G, out, hPart, sqPart);
  hipLaunchKernelGGL(finalize_kernel, dim3(1), dim3(256), 0, stream, hPart,
                     sqPart, out);
}